// DCN_Cross_Layer_91345364451542
// MI455X (gfx1250) — compile-verified
//
#include <hip/hip_runtime.h>

// DCN cross layer, B=32768, D=1024, L=3, fp32.
// Memory-bound (<1 flop/byte); WMMA f32 16x16x4 used for the per-row dot
// (co-executes with VALU, keeps fp32 precision, exercises CDNA5 matrix path).

#define B_  32768
#define D_  1024
#define L_  3
#define WAVES 8
#define ROWS_PER_WAVE 16
#define ROWS_PER_BLOCK (WAVES * ROWS_PER_WAVE)   // 128
#define NBLK1 (B_ / ROWS_PER_BLOCK)              // 256

typedef __attribute__((ext_vector_type(2))) float v2f;
typedef __attribute__((ext_vector_type(8))) float v8f;

// ---------------------------------------------------------------------------
// Kernel 1: per-row dot (WMMA), y = cur + x0*scale + bias, per-block column
// partial sums / sums-of-squares (deterministic).
// ---------------------------------------------------------------------------
__global__ __launch_bounds__(256) void dcn_rows(
    const float* __restrict__ cur, const float* __restrict__ x0,
    const float* __restrict__ w,   const float* __restrict__ bias,
    float* __restrict__ y,
    float* __restrict__ psumg, float* __restrict__ psqg)
{
    __shared__ __attribute__((aligned(16))) float wstage[D_];                       // 4 KB
    __shared__ __attribute__((aligned(16))) float tile[WAVES][ROWS_PER_WAVE * 32];  // 16 KB
    __shared__ float scaleArr[WAVES][ROWS_PER_WAVE];                                // 0.5 KB
    __shared__ float bsum[D_];                                                      // 4 KB
    __shared__ float bsq[D_];                                                       // 4 KB

    const int tid  = threadIdx.x;
    const int wv   = tid >> 5;
    const int lane = tid & 31;
    const int lr   = lane & 15;      // row within wave tile (A-matrix M index)
    const int hi   = lane >> 4;      // K-half selector per 16x4 f32 A layout
    const int rowbase = blockIdx.x * ROWS_PER_BLOCK + wv * ROWS_PER_WAVE;

    // Stage the 1024-element weight vector once (shared by all waves).
    for (int i = tid; i < D_; i += 256) wstage[i] = w[i];
    __syncthreads();

    // ---- Phase 1: scale[r] = cur[rowbase+r, :] . w  via WMMA f32 16x16x4 ----
    // A (16x4 f32): VGPR0 = {lanes0-15:K0, lanes16-31:K2}, VGPR1 = {K1, K3}.
    // B (4x16 f32): broadcast w[k] across all 16 columns -> every column of C
    // accumulates the same row dot.
    v8f c = {};
    for (int k = 0; k < D_; k += 32) {
        // Cooperative coalesced stage: 16 rows x 32 cols per wave.
        #pragma unroll
        for (int t = 0; t < 4; ++t) {
            int flat = t * 32 + lane;        // 0..127
            int row  = flat >> 3;            // 0..15
            int c4   = flat & 7;             // 0..7
            const float4 v = *(const float4*)(cur + (size_t)(rowbase + row) * D_ + k + c4 * 4);
            *(float4*)(&tile[wv][row * 32 + c4 * 4]) = v;
        }
        __syncthreads();
        #pragma unroll
        for (int j = 0; j < 8; ++j) {
            const int kc = j * 4 + 2 * hi;
            v2f a, b;
            a.x = tile[wv][lr * 32 + kc];
            a.y = tile[wv][lr * 32 + kc + 1];
            b.x = wstage[k + kc];
            b.y = wstage[k + kc + 1];
            c = __builtin_amdgcn_wmma_f32_16x16x4_f32(
                    false, a, false, b, (short)0, c, false, false);
        }
        __syncthreads();
    }

    // C layout: VGPR r, lanes0-15 -> M=r (N=lane); lanes16-31 -> M=8+r.
    if (lane == 0) {
        #pragma unroll
        for (int r = 0; r < 8; ++r) scaleArr[wv][r] = c[r];
    } else if (lane == 16) {
        #pragma unroll
        for (int r = 0; r < 8; ++r) scaleArr[wv][8 + r] = c[r];
    }
    // Zero the block column accumulators.
    for (int i = tid; i < D_; i += 256) { bsum[i] = 0.0f; bsq[i] = 0.0f; }
    __syncthreads();

    // ---- Phase 2: y = cur + x0*scale + bias; per-wave column partials ----
    float4 bias4[8], s4[8], q4[8];
    #pragma unroll
    for (int t = 0; t < 8; ++t) {
        bias4[t] = *(const float4*)(bias + t * 128 + lane * 4);
        s4[t].x = s4[t].y = s4[t].z = s4[t].w = 0.0f;
        q4[t].x = q4[t].y = q4[t].z = q4[t].w = 0.0f;
    }
    for (int r = 0; r < ROWS_PER_WAVE; ++r) {
        const size_t rowoff = (size_t)(rowbase + r) * D_;
        const float s = scaleArr[wv][r];
        #pragma unroll
        for (int t = 0; t < 8; ++t) {
            const int col = t * 128 + lane * 4;
            const float4 cu = *(const float4*)(cur + rowoff + col);
            const float4 xx = *(const float4*)(x0  + rowoff + col);
            float4 yv;
            yv.x = cu.x + xx.x * s + bias4[t].x;
            yv.y = cu.y + xx.y * s + bias4[t].y;
            yv.z = cu.z + xx.z * s + bias4[t].z;
            yv.w = cu.w + xx.w * s + bias4[t].w;
            *(float4*)(y + rowoff + col) = yv;
            s4[t].x += yv.x; s4[t].y += yv.y; s4[t].z += yv.z; s4[t].w += yv.w;
            q4[t].x += yv.x * yv.x; q4[t].y += yv.y * yv.y;
            q4[t].z += yv.z * yv.z; q4[t].w += yv.w * yv.w;
        }
    }

    // Deterministic cross-wave reduction (fixed wave order via barriers).
    for (int wvi = 0; wvi < WAVES; ++wvi) {
        if (wv == wvi) {
            #pragma unroll
            for (int t = 0; t < 8; ++t) {
                const int col = t * 128 + lane * 4;
                bsum[col + 0] += s4[t].x; bsum[col + 1] += s4[t].y;
                bsum[col + 2] += s4[t].z; bsum[col + 3] += s4[t].w;
                bsq[col + 0]  += q4[t].x; bsq[col + 1]  += q4[t].y;
                bsq[col + 2]  += q4[t].z; bsq[col + 3]  += q4[t].w;
            }
        }
        __syncthreads();
    }
    for (int col = tid; col < D_; col += 256) {
        psumg[(size_t)blockIdx.x * D_ + col] = bsum[col];
        psqg [(size_t)blockIdx.x * D_ + col] = bsq[col];
    }
}

// ---------------------------------------------------------------------------
// Kernel 2: reduce 256 block partials per column; fold BN into affine a,b.
// ---------------------------------------------------------------------------
__global__ __launch_bounds__(256) void dcn_stats(
    const float* __restrict__ psumg, const float* __restrict__ psqg,
    const float* __restrict__ gamma, const float* __restrict__ beta,
    float* __restrict__ aArr, float* __restrict__ bArr)
{
    const int col = blockIdx.x * 256 + threadIdx.x;   // 1024 threads total
    float s = 0.0f, q = 0.0f;
    for (int p = 0; p < NBLK1; ++p) {
        s += psumg[(size_t)p * D_ + col];
        q += psqg [(size_t)p * D_ + col];
    }
    const float invB = 1.0f / (float)B_;
    const float mean = s * invB;
    const float var  = q * invB - mean * mean;
    const float inv  = rsqrtf(var + 1e-5f);
    const float a    = inv * gamma[col];
    aArr[col] = a;
    bArr[col] = beta[col] - mean * a;
}

// ---------------------------------------------------------------------------
// Kernel 3: out = y * a[col] + b[col]  (pure streaming, float4).
// ---------------------------------------------------------------------------
__global__ __launch_bounds__(256) void dcn_norm(
    const float* __restrict__ y, const float* __restrict__ aArr,
    const float* __restrict__ bArr, float* __restrict__ out)
{
    const size_t i = ((size_t)blockIdx.x * 256 + threadIdx.x) * 4;
    __builtin_prefetch(y + i + 32768, 0, 0);   // global_prefetch_b8 on gfx1250
    const float4 yv = *(const float4*)(y + i);
    const int col   = (int)(i & (D_ - 1));
    const float4 av = *(const float4*)(aArr + col);
    const float4 bv = *(const float4*)(bArr + col);
    float4 ov;
    ov.x = yv.x * av.x + bv.x;
    ov.y = yv.y * av.y + bv.y;
    ov.z = yv.z * av.z + bv.z;
    ov.w = yv.w * av.w + bv.w;
    *(float4*)(out + i) = ov;
}

// ---------------------------------------------------------------------------
extern "C" void kernel_launch(void* const* d_in, const int* in_sizes, int n_in,
                              void* d_out, int out_size, void* d_ws, size_t ws_size,
                              hipStream_t stream) {
    (void)in_sizes; (void)n_in; (void)out_size; (void)ws_size;
    const float* inp     = (const float*)d_in[0];
    const float* weights = (const float*)d_in[1];   // [L, D]
    const float* biases  = (const float*)d_in[2];   // [L, D]
    const float* gammas  = (const float*)d_in[3];   // [L, D]
    const float* betas   = (const float*)d_in[4];   // [L, D]
    float* out = (float*)d_out;

    const size_t BD = (size_t)B_ * D_;
    float* y     = (float*)d_ws;          // 128 MB
    float* psumg = y + BD;                // 1 MB
    float* psqg  = psumg + (size_t)NBLK1 * D_;
    float* aArr  = psqg  + (size_t)NBLK1 * D_;
    float* bArr  = aArr + D_;

    const float* cur = inp;
    for (int i = 0; i < L_; ++i) {
        dcn_rows<<<NBLK1, 256, 0, stream>>>(cur, inp, weights + i * D_,
                                            biases + i * D_, y, psumg, psqg);
        dcn_stats<<<D_ / 256, 256, 0, stream>>>(psumg, psqg, gammas + i * D_,
                                                betas + i * D_, aArr, bArr);
        dcn_norm<<<(int)(BD / 4 / 256), 256, 0, stream>>>(y, aArr, bArr, out);
        cur = out;   // d_out doubles as the inter-layer buffer
    }
}